// SpectralTemporalMolecularNet_21131239096959
// MI455X (gfx1250) — compile-verified
//
#include <hip/hip_runtime.h>
#include <hip/hip_bf16.h>

// ---------------- constants ----------------
static constexpr int BG   = 1024;          // graphs
static constexpr int NPG  = 64;            // nodes per graph
static constexpr int NN   = BG * NPG;      // total nodes = 65536
static constexpr int HH   = 256;           // hidden
static constexpr int NSC  = 4;             // scales
static constexpr int NLL  = 4;             // layers
static constexpr int IND  = 9;             // input dim
static constexpr int EE   = 16 * NN;       // edges = 1,048,576
static constexpr int HEADS= 8;
static constexpr int HD   = HH / HEADS;    // 32
static constexpr int CH   = HH / 2;        // 128
static constexpr int STAGES = 4;

typedef __attribute__((ext_vector_type(2))) float v2f;
typedef __attribute__((ext_vector_type(8))) float v8f;

// ---------------- generic WMMA f32 GEMM (K % 4 == 0, M % 32 == 0, J % 64 == 0) ----
// Out[m, out_col_off + j] = act( sum_k Aeff[m,k] * B(k,j) + bias[j] )
// Aeff = A (+ A2 if ADD2).  TRANSB==0: W stored [K x J] (x @ W).
// TRANSB==1: W stored [J x K] (x @ W.T, torch Linear).
// ACT: 0=none 1=relu 2=tanh.  One wave -> 32 rows x 64 cols (8 wmma accumulators).
template<int TRANSB, int ACT, bool ADD2>
__global__ __launch_bounds__(256)
void gemm_wmma(const float* __restrict__ A, const float* __restrict__ A2,
               const float* __restrict__ W, const float* __restrict__ bias,
               float* __restrict__ Out,
               int M, int K, int J, int out_ld, int out_col_off)
{
    constexpr int ROWT = 2;                  // 2 row tiles of 16
    int wave = (int)((blockIdx.x * blockDim.x + threadIdx.x) >> 5);
    int lane = threadIdx.x & 31;
    int colTiles = J >> 6;
    int totalWaves = (M / (16 * ROWT)) * colTiles;
    if (wave >= totalWaves) return;          // wave-uniform: EXEC stays all-ones

    int rowTile = wave / colTiles;
    int colTile = wave % colTiles;
    int r0 = rowTile * (16 * ROWT);
    int c0 = colTile << 6;
    int lm = lane & 15;
    int kh = (lane >> 4) << 1;               // 0 or 2 (A/B k-sub-offset)

    const float* Ar[ROWT];
    const float* A2r[ROWT];
    #pragma unroll
    for (int rt = 0; rt < ROWT; ++rt) {
        Ar[rt] = A + (size_t)(r0 + rt * 16 + lm) * K + kh;
        if (ADD2) A2r[rt] = A2 + (size_t)(r0 + rt * 16 + lm) * K + kh;
    }

    v8f acc[ROWT][4];
    #pragma unroll
    for (int rt = 0; rt < ROWT; ++rt)
        #pragma unroll
        for (int jt = 0; jt < 4; ++jt)
            #pragma unroll
            for (int e = 0; e < 8; ++e) acc[rt][jt][e] = 0.0f;

    for (int kk = 0; kk < K; kk += 4) {
        v2f a[ROWT];
        #pragma unroll
        for (int rt = 0; rt < ROWT; ++rt) {
            a[rt] = *(const v2f*)(Ar[rt] + kk);
            if (ADD2) {
                v2f a2 = *(const v2f*)(A2r[rt] + kk);
                a[rt][0] += a2[0];
                a[rt][1] += a2[1];
            }
        }
        v2f b[4];
        #pragma unroll
        for (int jt = 0; jt < 4; ++jt) {
            int jc = c0 + (jt << 4) + lm;
            if (TRANSB) {
                b[jt] = *(const v2f*)(W + (size_t)jc * K + kk + kh);
            } else {
                b[jt][0] = W[(size_t)(kk + kh) * J + jc];
                b[jt][1] = W[(size_t)(kk + kh + 1) * J + jc];
            }
        }
        #pragma unroll
        for (int rt = 0; rt < ROWT; ++rt)
            #pragma unroll
            for (int jt = 0; jt < 4; ++jt)
                acc[rt][jt] = __builtin_amdgcn_wmma_f32_16x16x4_f32(
                    false, a[rt], false, b[jt], (short)0, acc[rt][jt], false, false);
    }

    int khi = lane >> 4;                     // 0/1 -> +8 row offset in D
    #pragma unroll
    for (int rt = 0; rt < ROWT; ++rt) {
        #pragma unroll
        for (int jt = 0; jt < 4; ++jt) {
            int jc = c0 + (jt << 4) + lm;
            float bv = bias ? bias[jc] : 0.0f;
            #pragma unroll
            for (int r = 0; r < 8; ++r) {
                int orow = r0 + rt * 16 + r + (khi << 3);
                float v = acc[rt][jt][r] + bv;
                if (ACT == 1) v = fmaxf(v, 0.0f);
                else if (ACT == 2) v = tanhf(v);
                Out[(size_t)orow * out_ld + out_col_off + jc] = v;
            }
        }
    }
}

// ---------------- K=9 projections (memory-bound; plain VALU) ----------------
template<int ACT>
__global__ __launch_bounds__(256)
void proj9(const float* __restrict__ X, const float* __restrict__ W,
           const float* __restrict__ bias, float* __restrict__ Out, int M)
{
    size_t idx = (size_t)blockIdx.x * blockDim.x + threadIdx.x;
    if (idx >= (size_t)M * HH) return;
    int n = (int)(idx >> 8);
    int j = (int)(idx & 255);
    float acc = bias[j];
    #pragma unroll
    for (int k = 0; k < IND; ++k) acc += X[(size_t)n * IND + k] * W[k * HH + j];
    if (ACT == 1) acc = fmaxf(acc, 0.0f);
    Out[idx] = acc;
}

// ---------------- fused per-(graph, head) attention ----------------
__global__ __launch_bounds__(64)
void attn_core(const float* __restrict__ qkv, float* __restrict__ out)
{
    int b  = blockIdx.x;
    int hh = blockIdx.y;
    int tid = threadIdx.x;                   // query row 0..63
    __shared__ float Ks[64][33];
    __shared__ float Vs[64][33];
    const float* base = qkv + (size_t)b * NPG * (3 * HH);
    const float* kp = base + (size_t)tid * (3 * HH) + HH     + hh * HD;
    const float* vp = base + (size_t)tid * (3 * HH) + 2 * HH + hh * HD;
    #pragma unroll
    for (int d = 0; d < HD; ++d) { Ks[tid][d] = kp[d]; Vs[tid][d] = vp[d]; }
    __syncthreads();

    float q[HD];
    const float* qp = base + (size_t)tid * (3 * HH) + hh * HD;
    #pragma unroll
    for (int d = 0; d < HD; ++d) q[d] = qp[d];

    const float scale = 0.17677669529663687f;  // 1/sqrt(32)
    float sc[NPG];
    float mx = -3.4e38f;
    for (int j = 0; j < NPG; ++j) {
        float s = 0.0f;
        #pragma unroll
        for (int d = 0; d < HD; ++d) s += q[d] * Ks[j][d];
        s *= scale;
        sc[j] = s;
        mx = fmaxf(mx, s);
    }
    float se = 0.0f;
    for (int j = 0; j < NPG; ++j) { sc[j] = __expf(sc[j] - mx); se += sc[j]; }
    float inv = 1.0f / se;
    float o[HD];
    #pragma unroll
    for (int d = 0; d < HD; ++d) o[d] = 0.0f;
    for (int j = 0; j < NPG; ++j) {
        float a = sc[j] * inv;
        #pragma unroll
        for (int d = 0; d < HD; ++d) o[d] += a * Vs[j][d];
    }
    float* op = out + ((size_t)b * NPG + tid) * HH + hh * HD;
    #pragma unroll
    for (int d = 0; d < HD; ++d) op[d] = o[d];
}

// ---------------- BN: partial-sum reduce + apply ----------------
__global__ void bn_reduce(const float* __restrict__ X, float* __restrict__ sums,
                          int rows, int feats, int rpb)
{
    int f  = threadIdx.x;                    // blockDim.x == feats
    int r0 = blockIdx.x * rpb;
    int r1 = r0 + rpb; if (r1 > rows) r1 = rows;
    float s = 0.0f, s2 = 0.0f;
    for (int r = r0; r < r1; ++r) {
        float v = X[(size_t)r * feats + f];
        s += v; s2 += v * v;
    }
    atomicAdd(&sums[f], s);
    atomicAdd(&sums[feats + f], s2);
}

__global__ void bn_apply(const float* __restrict__ X, const float* __restrict__ sums,
                         const float* __restrict__ gamma, const float* __restrict__ beta,
                         float* __restrict__ Y, int rows, int feats, int do_relu)
{
    size_t idx = (size_t)blockIdx.x * blockDim.x + threadIdx.x;
    size_t total = (size_t)rows * feats;
    if (idx >= total) return;
    int f = (int)(idx % feats);
    float invr = 1.0f / (float)rows;
    float m   = sums[f] * invr;
    float var = sums[feats + f] * invr - m * m;
    float v = (X[idx] - m) * rsqrtf(var + 1e-5f) * gamma[f] + beta[f];
    if (do_relu) v = fmaxf(v, 0.0f);
    Y[idx] = v;
}

// ---------------- residual add3 + LayerNorm (per row, 256 thr) ----------------
__global__ __launch_bounds__(256)
void add3_ln(const float* __restrict__ A, const float* __restrict__ Bm,
             const float* __restrict__ C, const float* __restrict__ gamma,
             const float* __restrict__ beta, float* __restrict__ Out)
{
    int row = blockIdx.x;
    int t = threadIdx.x;
    size_t idx = (size_t)row * HH + t;
    float v = A[idx] + Bm[idx] + C[idx];
    __shared__ float red[HH];
    __shared__ float mean_s, var_s;
    red[t] = v; __syncthreads();
    for (int s = HH / 2; s > 0; s >>= 1) {
        if (t < s) red[t] += red[t + s];
        __syncthreads();
    }
    if (t == 0) mean_s = red[0] * (1.0f / HH);
    __syncthreads();
    float d = v - mean_s;
    red[t] = d * d; __syncthreads();
    for (int s = HH / 2; s > 0; s >>= 1) {
        if (t < s) red[t] += red[t + s];
        __syncthreads();
    }
    if (t == 0) var_s = red[0] * (1.0f / HH);
    __syncthreads();
    Out[idx] = d * rsqrtf(var_s + 1e-5f) * gamma[t] + beta[t];
}

// ---------------- GIN edge scatter: agg[dst] += h[src] ----------------
__global__ void gin_agg(const float* __restrict__ h, const int* __restrict__ src,
                        const int* __restrict__ dst, float* __restrict__ agg)
{
    size_t tid = (size_t)blockIdx.x * blockDim.x + threadIdx.x;
    size_t total = (size_t)EE * 64;
    if (tid >= total) return;
    int e = (int)(tid >> 6);
    int c = (int)(tid & 63) * 4;
    const float4 v = *(const float4*)(h + (size_t)src[e] * HH + c);
    float* p = agg + (size_t)dst[e] * HH + c;
    atomicAdd(p + 0, v.x);
    atomicAdd(p + 1, v.y);
    atomicAdd(p + 2, v.z);
    atomicAdd(p + 3, v.w);
}

// ---------------- pooling ----------------
__global__ void pool_score(const float* __restrict__ tbuf, const float* __restrict__ W2,
                           const float* __restrict__ b2, float* __restrict__ s)
{
    int n = blockIdx.x * blockDim.x + threadIdx.x;
    if (n >= NN) return;
    float acc = b2[0];
    for (int k = 0; k < CH; ++k) acc += tbuf[(size_t)n * CH + k] * W2[k];
    s[n] = acc;
}

__global__ __launch_bounds__(256)
void pool_combine(const float* __restrict__ s, const float* __restrict__ h,
                  float* __restrict__ g)
{
    int b = blockIdx.x;
    int t = threadIdx.x;
    __shared__ float sc[NPG];
    if (t < NPG) sc[t] = s[b * NPG + t];
    __syncthreads();
    float mx = -3.4e38f;
    for (int n = 0; n < NPG; ++n) mx = fmaxf(mx, sc[n]);
    float se = 0.0f;
    for (int n = 0; n < NPG; ++n) se += __expf(sc[n] - mx);
    float inv = 1.0f / se;
    float acc = 0.0f;
    for (int n = 0; n < NPG; ++n)
        acc += __expf(sc[n] - mx) * inv * h[((size_t)b * NPG + n) * HH + t];
    g[(size_t)b * HH + t] = acc;
}

// ---------------- small epilogues ----------------
__global__ void cls_logits(const float* __restrict__ mid, const float* __restrict__ W2,
                           const float* __restrict__ b2, float* __restrict__ logits)
{
    int idx = blockIdx.x * blockDim.x + threadIdx.x;
    if (idx >= BG * STAGES) return;
    int b = idx >> 2, c = idx & 3;
    float acc = b2[c];
    for (int k = 0; k < CH; ++k) acc += mid[(size_t)b * CH + k] * W2[k * STAGES + c];
    logits[idx] = acc;
}

__global__ void head3(const float* __restrict__ t2, const float* __restrict__ W3,
                      const float* __restrict__ b3, float* __restrict__ pred)
{
    int b = blockIdx.x * blockDim.x + threadIdx.x;
    if (b >= BG) return;
    float acc = b3[0];
    for (int k = 0; k < CH / 2; ++k) acc += t2[(size_t)b * (CH / 2) + k] * W3[k];
    pred[b] = acc;
}

__global__ void finalize(const float* __restrict__ logits, const float* __restrict__ preds,
                         float* __restrict__ outPred)
{
    int b = blockIdx.x * blockDim.x + threadIdx.x;
    if (b >= BG) return;
    float l[STAGES];
    float mx = -3.4e38f;
    for (int c = 0; c < STAGES; ++c) { l[c] = logits[b * STAGES + c]; mx = fmaxf(mx, l[c]); }
    float se = 0.0f;
    for (int c = 0; c < STAGES; ++c) { l[c] = __expf(l[c] - mx); se += l[c]; }
    float inv = 1.0f / se;
    float p = 0.0f;
    for (int c = 0; c < STAGES; ++c) p += (l[c] * inv) * preds[c * BG + b];
    outPred[b] = p;
}

// ---------------- host launcher ----------------
extern "C" void kernel_launch(void* const* d_in, const int* in_sizes, int n_in,
                              void* d_out, int out_size, void* d_ws, size_t ws_size,
                              hipStream_t stream)
{
    (void)in_sizes; (void)n_in; (void)out_size; (void)ws_size;
    int i = 0;
    const float* x    = (const float*)d_in[i++];   // [N, 9]
    const int*   edge = (const int*)  d_in[i++];   // [2, E]
    i++;                                           // batch (unused)
    const float* sf   = (const float*)d_in[i++];   // [4, N, 9]
    const float* in_W = (const float*)d_in[i++];
    const float* in_b = (const float*)d_in[i++];
    const float* sp_W = (const float*)d_in[i++];
    const float* sp_b = (const float*)d_in[i++];
    struct Layer {
        const float *sc_W, *sc_b, *ai_W, *ai_b, *ao_W, *ao_b, *bn_g, *bn_b;
        const float *g1_W, *g1_b, *g1_bng, *g1_bnb, *g2_W, *g2_b, *ln_g, *ln_b;
    } L[NLL];
    for (int l = 0; l < NLL; ++l) {
        L[l].sc_W  = (const float*)d_in[i++]; L[l].sc_b  = (const float*)d_in[i++];
        L[l].ai_W  = (const float*)d_in[i++]; L[l].ai_b  = (const float*)d_in[i++];
        L[l].ao_W  = (const float*)d_in[i++]; L[l].ao_b  = (const float*)d_in[i++];
        L[l].bn_g  = (const float*)d_in[i++]; L[l].bn_b  = (const float*)d_in[i++];
        L[l].g1_W  = (const float*)d_in[i++]; L[l].g1_b  = (const float*)d_in[i++];
        L[l].g1_bng= (const float*)d_in[i++]; L[l].g1_bnb= (const float*)d_in[i++];
        L[l].g2_W  = (const float*)d_in[i++]; L[l].g2_b  = (const float*)d_in[i++];
        L[l].ln_g  = (const float*)d_in[i++]; L[l].ln_b  = (const float*)d_in[i++];
    }
    const float* pool_W1 = (const float*)d_in[i++];
    const float* pool_b1 = (const float*)d_in[i++];
    const float* pool_W2 = (const float*)d_in[i++];
    const float* pool_b2 = (const float*)d_in[i++];
    const float* cls_W1  = (const float*)d_in[i++];
    const float* cls_b1  = (const float*)d_in[i++];
    const float* cls_W2  = (const float*)d_in[i++];
    const float* cls_b2  = (const float*)d_in[i++];
    struct Head {
        const float *W1, *b1, *bng1, *bnb1, *W2, *b2, *bng2, *bnb2, *W3, *b3;
    } Hd[STAGES];
    for (int s = 0; s < STAGES; ++s) {
        Hd[s].W1   = (const float*)d_in[i++]; Hd[s].b1   = (const float*)d_in[i++];
        Hd[s].bng1 = (const float*)d_in[i++]; Hd[s].bnb1 = (const float*)d_in[i++];
        Hd[s].W2   = (const float*)d_in[i++]; Hd[s].b2   = (const float*)d_in[i++];
        Hd[s].bng2 = (const float*)d_in[i++]; Hd[s].bnb2 = (const float*)d_in[i++];
        Hd[s].W3   = (const float*)d_in[i++]; Hd[s].b3   = (const float*)d_in[i++];
    }

    const int* e_src = edge;
    const int* e_dst = edge + EE;

    // workspace partition (floats)
    const size_t NH = (size_t)NN * HH;
    float* W     = (float*)d_ws;
    float* h     = W;                 // NH
    float* psf   = W + NH;            // 4*NH
    float* hcat  = W + 5 * NH;        // NH   (also attn-out target, also pool tbuf)
    float* qkvb  = W + 6 * NH;        // 3*NH (also hgraph target: first NH)
    float* bufA  = W + 9 * NH;        // NH   (hb / hspec)
    float* bufB  = W + 10 * NH;       // NH   (gin mid)
    float* agg   = W + 11 * NH;       // NH
    float* small = W + 12 * NH;
    float* sums  = small;             // 512
    float* svec  = small + 512;       // NN
    float* gbuf  = svec + NN;         // BG*HH
    float* clsm  = gbuf + (size_t)BG * HH;       // BG*CH
    float* t1    = clsm + (size_t)BG * CH;       // BG*CH
    float* t2    = t1 + (size_t)BG * CH;         // BG*(CH/2)
    float* preds = t2 + (size_t)BG * (CH / 2);   // STAGES*BG

    float* outPred = (float*)d_out;              // [1024]
    float* outLog  = (float*)d_out + BG;         // [1024*4]

    auto gemm_blocks = [](int M, int J) {
        int waves = (M / 32) * (J / 64);
        return (waves * 32 + 255) / 256;
    };
    const int projBlocks = (int)(((size_t)NN * HH + 255) / 256);

    // 1) h = relu(x @ in_W + in_b)             [N,9]x[9,256]
    proj9<1><<<projBlocks, 256, 0, stream>>>(x, in_W, in_b, h, NN);

    // 2) proj_sf[s] = sf[s] @ sp_W + sp_b      [N,9]x[9,256]
    for (int s = 0; s < NSC; ++s)
        proj9<0><<<projBlocks, 256, 0, stream>>>(
            sf + (size_t)s * NN * IND, sp_W, sp_b, psf + (size_t)s * NH, NN);

    for (int l = 0; l < NLL; ++l) {
        // spectral per-scale linear with fused (h + proj_sf[s]); concat scale-major
        for (int s = 0; s < NSC; ++s)
            gemm_wmma<0, 0, true><<<gemm_blocks(NN, 64), 256, 0, stream>>>(
                h, psf + (size_t)s * NH,
                L[l].sc_W + (size_t)s * HH * 64, L[l].sc_b + s * 64,
                hcat, NN, HH, 64, HH, s * 64);

        // qkv = hcat @ attn_in_W.T + b          [N,256]x[768,256]^T
        gemm_wmma<1, 0, false><<<gemm_blocks(NN, 3 * HH), 256, 0, stream>>>(
            hcat, nullptr, L[l].ai_W, L[l].ai_b, qkvb, NN, HH, 3 * HH, 3 * HH, 0);

        // attention (writes concatenated heads back into hcat)
        attn_core<<<dim3(BG, HEADS), 64, 0, stream>>>(qkvb, hcat);

        // out projection -> bufA
        gemm_wmma<1, 0, false><<<gemm_blocks(NN, HH), 256, 0, stream>>>(
            hcat, nullptr, L[l].ao_W, L[l].ao_b, bufA, NN, HH, HH, HH, 0);

        // BN + relu (spectral path), in place
        hipMemsetAsync(sums, 0, 2 * HH * sizeof(float), stream);
        bn_reduce<<<NN / 256, HH, 0, stream>>>(bufA, sums, NN, HH, 256);
        bn_apply<<<(int)((NH + 255) / 256), 256, 0, stream>>>(
            bufA, sums, L[l].bn_g, L[l].bn_b, bufA, NN, HH, 1);

        // GIN: agg = segment_sum(h[src] -> dst)
        hipMemsetAsync(agg, 0, NH * sizeof(float), stream);
        {
            size_t thr = (size_t)EE * 64;
            gin_agg<<<(int)((thr + 255) / 256), 256, 0, stream>>>(h, e_src, e_dst, agg);
        }
        // z = h + agg ; bufB = z @ g1_W + b
        gemm_wmma<0, 0, true><<<gemm_blocks(NN, HH), 256, 0, stream>>>(
            h, agg, L[l].g1_W, L[l].g1_b, bufB, NN, HH, HH, HH, 0);
        hipMemsetAsync(sums, 0, 2 * HH * sizeof(float), stream);
        bn_reduce<<<NN / 256, HH, 0, stream>>>(bufB, sums, NN, HH, 256);
        bn_apply<<<(int)((NH + 255) / 256), 256, 0, stream>>>(
            bufB, sums, L[l].g1_bng, L[l].g1_bnb, bufB, NN, HH, 1);
        // hgraph (reuse qkv area) = bufB @ g2_W + b
        gemm_wmma<0, 0, false><<<gemm_blocks(NN, HH), 256, 0, stream>>>(
            bufB, nullptr, L[l].g2_W, L[l].g2_b, qkvb, NN, HH, HH, HH, 0);

        // h = LN(hspec + hgraph + h)
        add3_ln<<<NN, HH, 0, stream>>>(bufA, qkvb, h, L[l].ln_g, L[l].ln_b, h);
    }

    // attention pooling
    gemm_wmma<0, 2, false><<<gemm_blocks(NN, CH), 256, 0, stream>>>(
        h, nullptr, pool_W1, pool_b1, hcat, NN, HH, CH, CH, 0);      // tanh
    pool_score<<<NN / 256, 256, 0, stream>>>(hcat, pool_W2, pool_b2, svec);
    pool_combine<<<BG, HH, 0, stream>>>(svec, h, gbuf);

    // classifier (logits -> d_out tail)
    gemm_wmma<0, 1, false><<<gemm_blocks(BG, CH), 256, 0, stream>>>(
        gbuf, nullptr, cls_W1, cls_b1, clsm, BG, HH, CH, CH, 0);
    cls_logits<<<(BG * STAGES + 255) / 256, 256, 0, stream>>>(clsm, cls_W2, cls_b2, outLog);

    // stage heads
    for (int s = 0; s < STAGES; ++s) {
        gemm_wmma<0, 0, false><<<gemm_blocks(BG, CH), 256, 0, stream>>>(
            gbuf, nullptr, Hd[s].W1, Hd[s].b1, t1, BG, HH, CH, CH, 0);
        hipMemsetAsync(sums, 0, 2 * CH * sizeof(float), stream);
        bn_reduce<<<BG / 16, CH, 0, stream>>>(t1, sums, BG, CH, 16);
        bn_apply<<<(BG * CH + 255) / 256, 256, 0, stream>>>(
            t1, sums, Hd[s].bng1, Hd[s].bnb1, t1, BG, CH, 1);

        gemm_wmma<0, 0, false><<<gemm_blocks(BG, CH / 2), 256, 0, stream>>>(
            t1, nullptr, Hd[s].W2, Hd[s].b2, t2, BG, CH, CH / 2, CH / 2, 0);
        hipMemsetAsync(sums, 0, 2 * (CH / 2) * sizeof(float), stream);
        bn_reduce<<<BG / 16, CH / 2, 0, stream>>>(t2, sums, BG, CH / 2, 16);
        bn_apply<<<(BG * (CH / 2) + 255) / 256, 256, 0, stream>>>(
            t2, sums, Hd[s].bng2, Hd[s].bnb2, t2, BG, CH / 2, 1);

        head3<<<(BG + 255) / 256, 256, 0, stream>>>(t2, Hd[s].W3, Hd[s].b3, preds + s * BG);
    }

    finalize<<<(BG + 255) / 256, 256, 0, stream>>>(outLog, preds, outPred);
}